// ScaledDotProductAttention_62895501082943
// MI455X (gfx1250) — compile-verified
//
#include <hip/hip_runtime.h>

typedef __attribute__((ext_vector_type(2))) float v2f;
typedef __attribute__((ext_vector_type(4))) float v4f;
typedef __attribute__((ext_vector_type(8))) float v8f;
typedef __attribute__((ext_vector_type(4))) int   v4i;

#define B_ 4
#define H_ 16
#define S_ 1024
#define D_ 64

// ---- gfx1250 async global<->LDS path (guarded; falls back to direct copies) ----
#if __has_builtin(__builtin_amdgcn_global_load_async_to_lds_b128) && \
    __has_builtin(__builtin_amdgcn_s_wait_asynccnt)
#define HAVE_ASYNC_LD 1
#else
#define HAVE_ASYNC_LD 0
#endif

#if HAVE_ASYNC_LD && __has_builtin(__builtin_amdgcn_global_store_async_from_lds_b128)
#define HAVE_ASYNC_ST 1
#else
#define HAVE_ASYNC_ST 0
#endif

// Explicit address spaces so LDS traffic lowers to ds_* (not flat_*).
#define LDS_AS __attribute__((address_space(3)))
#define GLB_AS __attribute__((address_space(1)))
typedef LDS_AS float lds_f;
typedef LDS_AS v2f   lds_v2f;
typedef LDS_AS v4f   lds_v4f;
typedef GLB_AS v4i   g_v4i;
typedef LDS_AS v4i   l_v4i;

static __device__ __forceinline__ g_v4i* to_glb(const void* p) {
  return (g_v4i*)(unsigned long long)p;
}
static __device__ __forceinline__ l_v4i* to_lds(lds_f* p) {
  return (l_v4i*)p;
}

#if HAVE_ASYNC_LD
#define WAIT_ASYNC(n) __builtin_amdgcn_s_wait_asynccnt(n)
#else
#define WAIT_ASYNC(n) do { } while (0)
#endif

// Stage a 64x64 f32 chunk (16 KB) global -> LDS. 8 async b128 per thread.
static __device__ __forceinline__ void stage64x64(const float* g, lds_f* l, int tid) {
#if HAVE_ASYNC_LD
#pragma unroll
  for (int i = 0; i < 8; ++i)
    __builtin_amdgcn_global_load_async_to_lds_b128(
        to_glb(g + (tid + i * 128) * 4), to_lds(l + (tid + i * 128) * 4), 0, 0);
#else
#pragma unroll
  for (int i = 0; i < 8; ++i)
    ((lds_v4f*)l)[tid + i * 128] = ((const v4f*)g)[tid + i * 128];
#endif
}

// Stage a 16x64 f32 tile (4 KB) global -> LDS. 2 async b128 per thread.
static __device__ __forceinline__ void stage16x64(const float* g, lds_f* l, int tid) {
#if HAVE_ASYNC_LD
#pragma unroll
  for (int i = 0; i < 2; ++i)
    __builtin_amdgcn_global_load_async_to_lds_b128(
        to_glb(g + (tid + i * 128) * 4), to_lds(l + (tid + i * 128) * 4), 0, 0);
#else
#pragma unroll
  for (int i = 0; i < 2; ++i)
    ((lds_v4f*)l)[tid + i * 128] = ((const v4f*)g)[tid + i * 128];
#endif
}

// Stage the 16x64 bias sub-tile of chunk c (rows are S_-strided in global).
static __device__ __forceinline__ void stage_bias(const float* bg, int c,
                                                  lds_f* sB, int tid) {
#if HAVE_ASYNC_LD
#pragma unroll
  for (int i = 0; i < 2; ++i) {
    const int j    = tid + i * 128;   // v4f index 0..255
    const int row  = j >> 4;
    const int col4 = j & 15;
    __builtin_amdgcn_global_load_async_to_lds_b128(
        to_glb(bg + row * S_ + c * 64 + col4 * 4), to_lds(sB + j * 4), 0, 0);
  }
#else
#pragma unroll
  for (int i = 0; i < 2; ++i) {
    const int j    = tid + i * 128;
    const int row  = j >> 4;
    const int col4 = j & 15;
    ((lds_v4f*)sB)[j] = *(const v4f*)(bg + row * S_ + c * 64 + col4 * 4);
  }
#endif
}

// One 16x16 score tile: scale*Q@K^T + bias, masked -> sP.
static __device__ __forceinline__ void qk_chunk(
    int c, const lds_f* sK, const lds_f* sB, const v2f* aq,
    lds_f* sP, const int* __restrict__ mg, int wave, int lane)
{
  const int arow = lane & 15;
  const int koff = (lane >> 4) * 2;
  v8f acc = {};
#pragma unroll
  for (int s = 0; s < 16; ++s) {
    // B[kdim][n] = K[n][kdim]; K rows contiguous in d inside the LDS chunk.
    v2f bq = *(const lds_v2f*)(sK + (wave * 16 + arow) * D_ + 4 * s + koff);
    acc = __builtin_amdgcn_wmma_f32_16x16x4_f32(
        false, aq[s], false, bq, (short)0, acc, false, false);
  }
  const int col   = c * 64 + wave * 16 + arow;
  const int lcol  = wave * 16 + arow;            // column within the chunk
  const int mk    = mg[col];
  const int rbase = (lane >> 4) * 8;   // C/D: VGPR r -> row r + 8*(lane>=16)
#pragma unroll
  for (int r = 0; r < 8; ++r) {
    const int row = rbase + r;
    float sc = acc[r] + sB[row * 64 + lcol];
    sc = (mk == 0) ? -1e9f : sc;       // select keeps EXEC all-ones
    sP[row * S_ + col] = sc;
  }
}

// 16 WMMA steps of P @ V over one staged 64-row V chunk.
static __device__ __forceinline__ void pv_chunk(
    int c, const lds_f* sV, const lds_f* sP, v8f& oacc, int wave, int lane)
{
  const int arow = lane & 15;
  const int koff = (lane >> 4) * 2;
  const int n0   = wave * 16;
#pragma unroll
  for (int s = 0; s < 16; ++s) {
    const int kl = 4 * s + koff;
    v2f ap = *(const lds_v2f*)(sP + arow * S_ + c * 64 + kl);  // P fragment
    v2f bp;                                                    // V fragment
    bp.x = sV[kl * D_ + n0 + arow];
    bp.y = sV[(kl + 1) * D_ + n0 + arow];
    oacc = __builtin_amdgcn_wmma_f32_16x16x4_f32(
        false, ap, false, bp, (short)0, oacc, false, false);
  }
}

__global__ __launch_bounds__(128) void
ScaledDotProductAttention_62895501082943_kernel(
    const float* __restrict__ q, const float* __restrict__ k,
    const float* __restrict__ v, const float* __restrict__ bias,
    const int* __restrict__ mask, float* __restrict__ out,
    float* __restrict__ attn_out)
{
  extern __shared__ float smem[];
  // Rebase the dynamic-LDS pointer into addrspace(3): low 32 bits of a
  // generic 'shared' pointer are the LDS byte offset (aperture scheme).
  lds_f* sm   = (lds_f*)(unsigned int)(unsigned long long)&smem[0];
  lds_f* sQ   = sm;                // 16 x 64                    (1024 f)
  lds_f* sP   = sQ + 1024;         // 16 x 1024 scores -> probs  (16384 f)
  lds_f* sKV0 = sP + 16384;        // K/V staging buffer 0       (4096 f)
  lds_f* sKV1 = sKV0 + 4096;       // K/V staging buffer 1       (4096 f)
  lds_f* sB0  = sKV1 + 4096;       // bias staging buffer 0      (1024 f)
  lds_f* sB1  = sB0 + 1024;        // bias staging buffer 1      (1024 f)
  lds_f* sRed = sB1 + 1024;        // row-max partials           (128 f)
  lds_f* sRed2= sRed + 128;        // row-sum partials           (128 f)

  const int b  = blockIdx.z;
  const int h  = blockIdx.y;
  const int q0 = blockIdx.x * 16;
  const int tid  = threadIdx.x;
  const int wave = tid >> 5;
  const int lane = tid & 31;

  const float* qg = q    + (((size_t)(b * H_ + h) * S_ + q0) * D_);
  const float* kg = k    + ((size_t)(b * H_ + h) * S_ * D_);
  const float* vg = v    + ((size_t)(b * H_ + h) * S_ * D_);
  const float* bg = bias + ((size_t)b * S_ * S_ + (size_t)q0 * S_);
  const int*   mg = mask + (size_t)b * S_;
  float* og = out      + (((size_t)(b * H_ + h) * S_ + q0) * D_);
  float* ag = attn_out + (((size_t)(b * H_ + h) * S_ + q0) * S_);

  // ---- Prologue: stage Q, K chunk 0, bias chunk 0 (12 async/thread) ----
  stage16x64(qg, sQ, tid);         // 2
  stage64x64(kg, sKV0, tid);       // 8
  stage_bias(bg, 0, sB0, tid);     // 2
  WAIT_ASYNC(10);                  // Q retired (async loads retire in order)
  __syncthreads();

  // Preload scaled Q A-fragments (reused for all 64 score tiles).
  // lanes 0-15: M=lane, K={4s,4s+1}; lanes 16-31: M=lane-16, K={4s+2,4s+3}
  const int arow = lane & 15;
  const int koff = (lane >> 4) * 2;
  v2f aq[16];
#pragma unroll
  for (int s = 0; s < 16; ++s)
    aq[s] = (*(const lds_v2f*)(sQ + arow * D_ + 4 * s + koff)) * 0.125f;

  // ---- Phase 1: scores, 16 chunks of 64 K-rows, double-buffered DMA ----
  for (int c = 0; c < 15; ++c) {
    lds_f* kdst = ((c + 1) & 1) ? sKV1 : sKV0;
    lds_f* bdst = ((c + 1) & 1) ? sB1 : sB0;
    stage64x64(kg + (size_t)(c + 1) * 64 * D_, kdst, tid);  // 8
    stage_bias(bg, c + 1, bdst, tid);                       // 2
    WAIT_ASYNC(10);               // chunk c (K + bias) resident
    __syncthreads();
    qk_chunk(c, (c & 1) ? sKV1 : sKV0, (c & 1) ? sB1 : sB0,
             aq, sP, mg, wave, lane);
    __syncthreads();
  }
  WAIT_ASYNC(0);
  __syncthreads();
  qk_chunk(15, sKV1, sB1, aq, sP, mg, wave, lane);
  __syncthreads();

  // Kick off V staging for PV now: DMA overlaps the softmax below.
  stage64x64(vg, sKV0, tid);
  stage64x64(vg + 64 * D_, sKV1, tid);

  // ---- Phase 2: softmax over each of the 16 rows (8 lanes/row, v4f) ----
  {
    const int srow = tid >> 3;
    const int sseg = tid & 7;
    lds_v4f* rowp = (lds_v4f*)(sP + srow * S_ + sseg * 128);

    float lmax = -3.4e38f;
#pragma unroll 8
    for (int i = 0; i < 32; ++i) {
      v4f t = rowp[i];
      lmax = fmaxf(lmax, fmaxf(fmaxf(t.x, t.y), fmaxf(t.z, t.w)));
    }
    sRed[tid] = lmax;
    __syncthreads();
    float rmax = sRed[srow * 8];
#pragma unroll
    for (int j = 1; j < 8; ++j) rmax = fmaxf(rmax, sRed[srow * 8 + j]);

    float lsum = 0.0f;
#pragma unroll 8
    for (int i = 0; i < 32; ++i) {
      v4f t = rowp[i];
      t.x = __expf(t.x - rmax); t.y = __expf(t.y - rmax);
      t.z = __expf(t.z - rmax); t.w = __expf(t.w - rmax);
      rowp[i] = t;
      lsum += (t.x + t.y) + (t.z + t.w);
    }
    sRed2[tid] = lsum;
    __syncthreads();
    float rsum = 0.0f;
#pragma unroll
    for (int j = 0; j < 8; ++j) rsum += sRed2[srow * 8 + j];
    const float rinv = 1.0f / rsum;
#pragma unroll 8
    for (int i = 0; i < 32; ++i) {
      v4f t = rowp[i];
      rowp[i] = t * rinv;
    }
  }
  __syncthreads();

  // ---- Phase 3: out = P @ V, 16 chunks of 64 V-rows, double-buffered ----
  v8f oacc = {};
  for (int c = 0; c < 14; ++c) {
    WAIT_ASYNC(8);                // chunk c resident
    __syncthreads();
    pv_chunk(c, (c & 1) ? sKV1 : sKV0, sP, oacc, wave, lane);
    __syncthreads();
    stage64x64(vg + (size_t)(c + 2) * 64 * D_, (c & 1) ? sKV1 : sKV0, tid);
  }
  WAIT_ASYNC(8);
  __syncthreads();
  pv_chunk(14, sKV0, sP, oacc, wave, lane);
  WAIT_ASYNC(0);
  __syncthreads();
  pv_chunk(15, sKV1, sP, oacc, wave, lane);

  // Store the output tile.
  {
    const int n0 = wave * 16;
    const int n = lane & 15;
    const int rbase = (lane >> 4) * 8;
#pragma unroll
    for (int r = 0; r < 8; ++r)
      og[(size_t)(rbase + r) * D_ + n0 + n] = oacc[r];
  }

  // ---- Stream probabilities LDS -> global attn (async b128 stores) ----
#if HAVE_ASYNC_ST
#pragma unroll
  for (int i = 0; i < 32; ++i)
    __builtin_amdgcn_global_store_async_from_lds_b128(
        to_glb(ag + (tid + i * 128) * 4), to_lds(sP + (tid + i * 128) * 4), 0, 0);
  WAIT_ASYNC(0);
#else
  {
    v4f* ag4 = (v4f*)ag;
#pragma unroll
    for (int i = 0; i < 32; ++i) {
      v4f t = ((const lds_v4f*)sP)[tid + i * 128];
      ag4[tid + i * 128] = t;
    }
  }
#endif
}

extern "C" void kernel_launch(void* const* d_in, const int* in_sizes, int n_in,
                              void* d_out, int out_size, void* d_ws, size_t ws_size,
                              hipStream_t stream) {
  const float* q    = (const float*)d_in[0];
  const float* k    = (const float*)d_in[1];
  const float* v    = (const float*)d_in[2];
  const float* bias = (const float*)d_in[3];
  const int*   mask = (const int*)d_in[4];

  float* out  = (float*)d_out;
  float* attn = out + (size_t)B_ * H_ * S_ * D_;   // tuple: (output, attn)

  dim3 grid(S_ / 16, H_, B_);
  dim3 block(128);
  const size_t shmem =
      (1024 + 16384 + 4096 + 4096 + 1024 + 1024 + 128 + 128) * sizeof(float);

  hipLaunchKernelGGL(ScaledDotProductAttention_62895501082943_kernel,
                     grid, block, shmem, stream,
                     q, k, v, bias, mask, out, attn);
}